// GAT_72387378807000
// MI455X (gfx1250) — compile-verified
//
#include <hip/hip_runtime.h>

#define NEG_SLOPE 0.2f
#define TB 256

typedef __attribute__((ext_vector_type(2))) float v2f;
typedef __attribute__((ext_vector_type(8))) float v8f;

// ---------- order-preserving float<->uint for atomic max ----------
__device__ __forceinline__ unsigned f2ord(float f) {
  unsigned u = __float_as_uint(f);
  return (u & 0x80000000u) ? ~u : (u | 0x80000000u);
}
__device__ __forceinline__ float ord2f(unsigned u) {
  return __uint_as_float((u & 0x80000000u) ? (u & 0x7fffffffu) : ~u);
}

// ---------- init helpers ----------
__global__ void zero_f32(float* __restrict__ p, int n) {
  int i = blockIdx.x * blockDim.x + threadIdx.x;
  if (i < n) p[i] = 0.f;
}
__global__ void fill_u32(unsigned* __restrict__ p, unsigned v, int n) {
  int i = blockIdx.x * blockDim.x + threadIdx.x;
  if (i < n) p[i] = v;
}

// Build fused weights: W1[128][16] = [Wl1 | Wr1], b1[16]; W2[8][32] = [Wl2 | Wr2], b2[32]
__global__ void build_weights(const float* __restrict__ Wl1, const float* __restrict__ bl1,
                              const float* __restrict__ Wr1, const float* __restrict__ br1,
                              const float* __restrict__ Wl2, const float* __restrict__ bl2,
                              const float* __restrict__ Wr2, const float* __restrict__ br2,
                              float* __restrict__ W1, float* __restrict__ b1,
                              float* __restrict__ W2, float* __restrict__ b2) {
  int t = threadIdx.x;
  for (int i = t; i < 128 * 16; i += TB) {
    int k = i >> 4, c = i & 15;
    W1[i] = (c < 8) ? Wl1[k * 8 + c] : Wr1[k * 8 + (c - 8)];
  }
  if (t < 16) b1[t] = (t < 8) ? bl1[t] : br1[t - 8];
  for (int i = t; i < 8 * 32; i += TB) {
    int k = i >> 5, c = i & 31;
    W2[i] = (c < 16) ? Wl2[k * 16 + c] : Wr2[k * 16 + (c - 16)];
  }
  if (t < 32) b2[t] = (t < 16) ? bl2[t] : br2[t - 16];
}

// ---------- self-loop edge_attr mean (fill_value='mean') ----------
__global__ void edge_deg(const int* __restrict__ dst, const float* __restrict__ ea,
                         float* __restrict__ cnt, float* __restrict__ easum, int E) {
  int e = blockIdx.x * blockDim.x + threadIdx.x;
  if (e >= E) return;
  int d = dst[e];
  atomicAdd(&cnt[d], 1.f);
  atomicAdd(&easum[d], ea[e]);
}
__global__ void ealoop_k(const float* __restrict__ cnt, const float* __restrict__ easum,
                         float* __restrict__ ealoop, int n) {
  int i = blockIdx.x * blockDim.x + threadIdx.x;
  if (i < n) ealoop[i] = easum[i] / fmaxf(cnt[i], 1.f);
}

// ---------- fused [lin_l | lin_r] projection via f32 WMMA ----------
// O[n][NO] = (RELU_IN ? relu(X) : X)[n][K] @ W[K][NO] + bias[NO]
// One wave = one 16-row tile; B fragments preloaded into registers; A-row index is
// clamped (not predicated) so the hot loop has no branches / EXEC churn.
template <int K, int NO, bool RELU_IN>
__global__ void gemm_wmma(const float* __restrict__ X, const float* __restrict__ W,
                          const float* __restrict__ bias, float* __restrict__ O, int n) {
  static_assert(K % 4 == 0 && NO % 16 == 0, "tile shape");
  constexpr int NT = NO / 16;
  constexpr int KS = K / 4;
  int wave = blockIdx.x * (blockDim.x >> 5) + (threadIdx.x >> 5);
  int ntiles = (n + 15) >> 4;
  if (wave >= ntiles) return;           // wave-uniform: EXEC stays all-1s for WMMA
  int lane = threadIdx.x & 31;
  int m = lane & 15;                    // row (A) / column (B,D) within tile
  int kh = lane >> 4;                   // K-half select (lanes 16-31: K+2)
  int r0 = wave << 4;
  int row = r0 + m;
  row = row < n ? row : n - 1;          // clamp: duplicate rows computed, never stored
  const float* xrow = X + (size_t)row * K + 2 * kh;
  const float* wcol = W + (size_t)(2 * kh) * NO + m;

  // Preload all B fragments (weights <= 8KB, L0/L2-hot)
  v2f bfrag[KS][NT];
#pragma unroll
  for (int ks = 0; ks < KS; ++ks) {
#pragma unroll
    for (int t = 0; t < NT; ++t) {
      bfrag[ks][t].x = wcol[(size_t)(4 * ks) * NO + t * 16];
      bfrag[ks][t].y = wcol[(size_t)(4 * ks + 1) * NO + t * 16];
    }
  }

  v8f acc[NT] = {};
#pragma unroll
  for (int ks = 0; ks < KS; ++ks) {
    v2f a = *(const v2f*)(xrow + 4 * ks);
    if (RELU_IN) { a.x = fmaxf(a.x, 0.f); a.y = fmaxf(a.y, 0.f); }
#pragma unroll
    for (int t = 0; t < NT; ++t)
      acc[t] = __builtin_amdgcn_wmma_f32_16x16x4_f32(false, a, false, bfrag[ks][t],
                                                     (short)0, acc[t], false, false);
  }

#pragma unroll
  for (int r = 0; r < 8; ++r) {
    int ro = r0 + r + 8 * kh;           // D layout: lanes16-31 hold M=r+8
    if (ro < n) {
#pragma unroll
      for (int t = 0; t < NT; ++t)
        O[(size_t)ro * NO + t * 16 + m] = acc[t][r] + bias[t * 16 + m];
    }
  }
}

// ---------- GATv2 edge/node kernels ----------
template <int H>
__device__ __forceinline__ float gat_logit(const float* __restrict__ xl,
                                           const float* __restrict__ xr, float eav,
                                           const float* __restrict__ We,
                                           const float* __restrict__ att) {
  float logit = 0.f;
#pragma unroll
  for (int h = 0; h < H; h += 4) {
    float4 a = *(const float4*)(xl + h);
    float4 b = *(const float4*)(xr + h);
    float4 w = *(const float4*)(We + h);
    float4 t = *(const float4*)(att + h);
    float m0 = a.x + b.x + eav * w.x; m0 = m0 > 0.f ? m0 : NEG_SLOPE * m0;
    float m1 = a.y + b.y + eav * w.y; m1 = m1 > 0.f ? m1 : NEG_SLOPE * m1;
    float m2 = a.z + b.z + eav * w.z; m2 = m2 > 0.f ? m2 : NEG_SLOPE * m2;
    float m3 = a.w + b.w + eav * w.w; m3 = m3 > 0.f ? m3 : NEG_SLOPE * m3;
    logit += m0 * t.x + m1 * t.y + m2 * t.z + m3 * t.w;
  }
  return logit;
}

// Pass A: per-edge logit, store it, atomic segment-max over dst.
template <int H, int XLS>
__global__ void edge_logit_max(const int* __restrict__ src, const int* __restrict__ dst,
                               const float* __restrict__ ea, const float* __restrict__ xlr,
                               const float* __restrict__ We, const float* __restrict__ att,
                               float* __restrict__ logits, unsigned* __restrict__ mx, int E) {
  int e = blockIdx.x * blockDim.x + threadIdx.x;
  if (e >= E) return;
  int s = src[e], d = dst[e];
  float l = gat_logit<H>(xlr + (size_t)s * XLS, xlr + (size_t)d * XLS + H, ea[e], We, att);
  logits[e] = l;
  atomicMax(mx + d, f2ord(l));
}

// Node pass A: self-loop logit, finalize max (stored as float bits in mx), seed denom.
template <int H, int XLS>
__global__ void node_loop_max(const float* __restrict__ xlr, const float* __restrict__ ealoop,
                              const float* __restrict__ We, const float* __restrict__ att,
                              unsigned* __restrict__ mx, float* __restrict__ lloop,
                              float* __restrict__ denom, int n) {
  int i = blockIdx.x * blockDim.x + threadIdx.x;
  if (i >= n) return;
  const float* xl = xlr + (size_t)i * XLS;
  float l = gat_logit<H>(xl, xl + H, ealoop[i], We, att);
  float mxv = fmaxf(ord2f(mx[i]), l);
  mx[i] = __float_as_uint(mxv);
  lloop[i] = l;
  denom[i] = expf(l - mxv);           // self-loop term seeds the softmax denominator
}

// Pass B: ex = exp(logit - mx[dst]); overwrite logits with ex; accumulate denom.
__global__ void edge_exp_sum(const int* __restrict__ dst, const unsigned* __restrict__ mx,
                             float* __restrict__ logits, float* __restrict__ denom, int E) {
  int e = blockIdx.x * blockDim.x + threadIdx.x;
  if (e >= E) return;
  int d = dst[e];
  float ex = expf(logits[e] - __uint_as_float(mx[d]));
  logits[e] = ex;
  atomicAdd(&denom[d], ex);
}

// Node pass B: out = bias + alpha_selfloop * xl[i]  (initializes out before edge scatter)
template <int H, int XLS>
__global__ void node_init_out(const float* __restrict__ xlr, const unsigned* __restrict__ mx,
                              const float* __restrict__ lloop, const float* __restrict__ denom,
                              const float* __restrict__ bias, float* __restrict__ out, int n) {
  int i = blockIdx.x * blockDim.x + threadIdx.x;
  if (i >= n) return;
  float alpha = expf(lloop[i] - __uint_as_float(mx[i])) / denom[i];
  const float* xl = xlr + (size_t)i * XLS;
#pragma unroll
  for (int h = 0; h < H; ++h) out[(size_t)i * H + h] = bias[h] + alpha * xl[h];
}

// Pass C: out[dst] += (ex/denom[dst]) * xl[src]
template <int H, int XLS>
__global__ void edge_scatter(const int* __restrict__ src, const int* __restrict__ dst,
                             const float* __restrict__ logits, const float* __restrict__ denom,
                             const float* __restrict__ xlr, float* __restrict__ out, int E) {
  int e = blockIdx.x * blockDim.x + threadIdx.x;
  if (e >= E) return;
  int s = src[e], d = dst[e];
  float alpha = logits[e] / denom[d];
  const float* xl = xlr + (size_t)s * XLS;
  float* o = out + (size_t)d * H;
#pragma unroll
  for (int h = 0; h < H; ++h) atomicAdd(o + h, alpha * xl[h]);
}

// ---------- pooling + final linear ----------
__global__ void pool_k(const float* __restrict__ h2, const int* __restrict__ batch,
                       float* __restrict__ pooled, float* __restrict__ gcnt, int n) {
  int i = blockIdx.x * blockDim.x + threadIdx.x;
  if (i >= n) return;
  int b = batch[i];
  atomicAdd(&gcnt[b], 1.f);
  const float* hp = h2 + (size_t)i * 16;
  float* pp = pooled + (size_t)b * 16;
#pragma unroll
  for (int h = 0; h < 16; ++h) atomicAdd(pp + h, fmaxf(hp[h], 0.f));  // relu folded in
}

__global__ void final_k(const float* __restrict__ pooled, const float* __restrict__ gcnt,
                        const float* __restrict__ W3, const float* __restrict__ b3,
                        float* __restrict__ out, int B) {
  int b = blockIdx.x * blockDim.x + threadIdx.x;
  if (b >= B) return;
  float inv = 1.f / fmaxf(gcnt[b], 1.f);
  float acc = b3[0];
#pragma unroll
  for (int h = 0; h < 16; ++h) acc += pooled[(size_t)b * 16 + h] * inv * W3[h];
  out[b] = acc;
}

extern "C" void kernel_launch(void* const* d_in, const int* in_sizes, int n_in,
                              void* d_out, int out_size, void* d_ws, size_t ws_size,
                              hipStream_t stream) {
  const float* x     = (const float*)d_in[0];
  const int*   ei    = (const int*)d_in[1];
  const float* ea    = (const float*)d_in[2];
  const int*   batch = (const int*)d_in[3];
  const float* Wl1 = (const float*)d_in[4],  *bl1 = (const float*)d_in[5];
  const float* Wr1 = (const float*)d_in[6],  *br1 = (const float*)d_in[7];
  const float* We1 = (const float*)d_in[8],  *att1 = (const float*)d_in[9];
  const float* bias1 = (const float*)d_in[10];
  const float* Wl2 = (const float*)d_in[11], *bl2 = (const float*)d_in[12];
  const float* Wr2 = (const float*)d_in[13], *br2 = (const float*)d_in[14];
  const float* We2 = (const float*)d_in[15], *att2 = (const float*)d_in[16];
  const float* bias2 = (const float*)d_in[17];
  const float* W3 = (const float*)d_in[18],  *b3 = (const float*)d_in[19];
  (void)n_in; (void)ws_size;

  const int n = in_sizes[0] / 128;   // N nodes
  const int E = in_sizes[2];         // edge_attr is (E,1)
  const int B = out_size;            // graphs
  const int* src = ei;
  const int* dst = ei + E;

  // workspace carve-out (256B-aligned slabs)
  char* ws = (char*)d_ws;
  size_t off = 0;
  auto alloc = [&](size_t elems) -> float* {
    float* p = (float*)(ws + off);
    off += ((elems * sizeof(float)) + 255) & ~(size_t)255;
    return p;
  };
  float*    W1     = alloc(128 * 16);
  float*    b1     = alloc(16);
  float*    W2     = alloc(8 * 32);
  float*    b2     = alloc(32);
  float*    xlr1   = alloc((size_t)n * 16);   // [xl1 | xr1]
  float*    out1   = alloc((size_t)n * 8);
  float*    xlr2   = alloc((size_t)n * 32);   // [xl2 | xr2]
  float*    out2   = alloc((size_t)n * 16);
  float*    logits = alloc(E);                // reused: logit -> ex
  float*    cnt    = alloc(n);
  float*    easum  = alloc(n);
  float*    ealoop = alloc(n);
  unsigned* mx     = (unsigned*)alloc(n);     // ordered-uint max, then float max
  float*    lloop  = alloc(n);
  float*    denom  = alloc(n);
  float*    pooled = alloc((size_t)B * 16);
  float*    gcnt   = alloc(B);

  const int gN = (n + TB - 1) / TB;
  const int gE = (E + TB - 1) / TB;
  const int ntiles = (n + 15) / 16;
  const int gW = (ntiles + 7) / 8;            // 8 waves (wave32) per 256-thread block
  const unsigned NEG_INF_ORD = 0x007fffffu;   // f2ord(-inf)

  build_weights<<<1, TB, 0, stream>>>(Wl1, bl1, Wr1, br1, Wl2, bl2, Wr2, br2, W1, b1, W2, b2);

  // self-loop edge_attr = per-destination mean of incoming attrs
  zero_f32<<<gN, TB, 0, stream>>>(cnt, n);
  zero_f32<<<gN, TB, 0, stream>>>(easum, n);
  edge_deg<<<gE, TB, 0, stream>>>(dst, ea, cnt, easum, E);
  ealoop_k<<<gN, TB, 0, stream>>>(cnt, easum, ealoop, n);

  // ---- layer 1 ----
  gemm_wmma<128, 16, false><<<gW, TB, 0, stream>>>(x, W1, b1, xlr1, n);
  fill_u32<<<gN, TB, 0, stream>>>(mx, NEG_INF_ORD, n);
  edge_logit_max<8, 16><<<gE, TB, 0, stream>>>(src, dst, ea, xlr1, We1, att1, logits, mx, E);
  node_loop_max<8, 16><<<gN, TB, 0, stream>>>(xlr1, ealoop, We1, att1, mx, lloop, denom, n);
  edge_exp_sum<<<gE, TB, 0, stream>>>(dst, mx, logits, denom, E);
  node_init_out<8, 16><<<gN, TB, 0, stream>>>(xlr1, mx, lloop, denom, bias1, out1, n);
  edge_scatter<8, 16><<<gE, TB, 0, stream>>>(src, dst, logits, denom, xlr1, out1, E);

  // ---- layer 2 (relu of layer-1 output folded into GEMM input load) ----
  gemm_wmma<8, 32, true><<<gW, TB, 0, stream>>>(out1, W2, b2, xlr2, n);
  fill_u32<<<gN, TB, 0, stream>>>(mx, NEG_INF_ORD, n);
  edge_logit_max<16, 32><<<gE, TB, 0, stream>>>(src, dst, ea, xlr2, We2, att2, logits, mx, E);
  node_loop_max<16, 32><<<gN, TB, 0, stream>>>(xlr2, ealoop, We2, att2, mx, lloop, denom, n);
  edge_exp_sum<<<gE, TB, 0, stream>>>(dst, mx, logits, denom, E);
  node_init_out<16, 32><<<gN, TB, 0, stream>>>(xlr2, mx, lloop, denom, bias2, out2, n);
  edge_scatter<16, 32><<<gE, TB, 0, stream>>>(src, dst, logits, denom, xlr2, out2, E);

  // ---- pooling + final linear (relu folded into pool read) ----
  zero_f32<<<(B * 16 + TB - 1) / TB, TB, 0, stream>>>(pooled, B * 16);
  zero_f32<<<(B + TB - 1) / TB, TB, 0, stream>>>(gcnt, B);
  pool_k<<<gN, TB, 0, stream>>>(out2, batch, pooled, gcnt, n);
  final_k<<<(B + TB - 1) / TB, TB, 0, stream>>>(pooled, gcnt, W3, b3, (float*)d_out, B);
}